// GNNLSTM_32238024524350
// MI455X (gfx1250) — compile-verified
//
#include <hip/hip_runtime.h>
#include <hip/hip_bf16.h>
#include <math.h>
#include <stdint.h>

typedef float v2f __attribute__((ext_vector_type(2)));
typedef float v8f __attribute__((ext_vector_type(8)));

#define B_    4
#define S_    32
#define N_    200
#define FIN_  32
#define E_    3200
#define GH_   64
#define HEADS_ 4
#define HC_   256      // HEADS_*GH_
#define LH_   128
#define TG_   (B_*S_)        // 128
#define NT_   (TG_*N_)       // 25600
#define TGE_  (TG_*E_)       // 409600
#define EA_   (TGE_+NT_)     // 435200
#define NEGS_ 0.2f
#define LN_EPS_ 1e-5f

__device__ __forceinline__ float sigm(float x){ return 1.f/(1.f+__expf(-x)); }

// ---------------- fill ----------------
__global__ void fill_kernel(float* p, float v, size_t n){
  size_t i = (size_t)blockIdx.x*blockDim.x + threadIdx.x;
  size_t st = (size_t)gridDim.x*blockDim.x;
  for (; i < n; i += st) p[i] = v;
}

// ---------------- generic WMMA f32 GEMM ----------------
// C[M,N] = A[M,K] @ B  (+bias +bias2 per column)
// BT=false: B stored row-major [K,N].  BT=true: B stored [N,K] (use B^T).
// One wave per 16x32 output strip (two 16x16 WMMA tiles sharing the A fragment).
template<bool BT>
__global__ void wmma_gemm_f32(const float* __restrict__ A, const float* __restrict__ Bm,
                              const float* __restrict__ bias, const float* __restrict__ bias2,
                              float* __restrict__ C, int M, int N, int K)
{
  int wave = (int)((blockIdx.x*blockDim.x + threadIdx.x) >> 5);
  int lane = threadIdx.x & 31;
  int ntiles2 = N >> 5;                   // strips of 32 columns
  int tM = wave / ntiles2;
  int tS = wave - tM*ntiles2;
  if (tM*16 >= M) return;                 // wave-uniform: EXEC stays all-ones
  int m  = lane & 15;                     // A row within tile
  int kb = (lane >> 4) << 1;              // K sub-offset (0 or 2)
  int n  = lane & 15;                     // B/C col within tile
  int mb = (lane >> 4) << 3;              // C row base (0 or 8)
  int col0 = tS*32 + n;
  int col1 = col0 + 16;
  const float* arow = A + (size_t)(tM*16 + m)*K;
  v8f acc0 = {}, acc1 = {};
  for (int k = 0; k < K; k += 4) {
    v2f a, b0, b1;
    a[0] = arow[k+kb]; a[1] = arow[k+kb+1];
    if (BT) {
      const float* bp0 = Bm + (size_t)col0*K + k + kb;
      const float* bp1 = Bm + (size_t)col1*K + k + kb;
      b0[0] = bp0[0]; b0[1] = bp0[1];
      b1[0] = bp1[0]; b1[1] = bp1[1];
    } else {
      b0[0] = Bm[(size_t)(k+kb  )*N + col0];
      b0[1] = Bm[(size_t)(k+kb+1)*N + col0];
      b1[0] = Bm[(size_t)(k+kb  )*N + col1];
      b1[1] = Bm[(size_t)(k+kb+1)*N + col1];
    }
    acc0 = __builtin_amdgcn_wmma_f32_16x16x4_f32(false, a, false, b0, (short)0, acc0, false, false);
    acc1 = __builtin_amdgcn_wmma_f32_16x16x4_f32(false, a, false, b1, (short)0, acc1, false, false);
  }
  float bv0 = 0.f, bv1 = 0.f;
  if (bias)  { bv0 += bias[col0];  bv1 += bias[col1];  }
  if (bias2) { bv0 += bias2[col0]; bv1 += bias2[col1]; }
  for (int v = 0; v < 8; ++v) {
    size_t r = (size_t)(tM*16 + mb + v)*N;
    C[r + col0] = acc0[v] + bv0;
    C[r + col1] = acc1[v] + bv1;
  }
}

// ---------------- GATv2 edge passes ----------------
__device__ __forceinline__ void edge_decode(int e, const int* __restrict__ ei, int& s, int& d){
  if (e < TGE_) {
    int g = e / E_;
    int j = e - g*E_;
    s = ei[j]      + g*N_;
    d = ei[E_ + j] + g*N_;
  } else {
    s = d = e - TGE_;   // self loop
  }
}

__device__ __forceinline__ void atomicMaxF(float* addr, float val){
  unsigned int* ua = (unsigned int*)addr;
  unsigned int old = *ua;
  while (__uint_as_float(old) < val) {
    unsigned int assumed = old;
    old = atomicCAS(ua, assumed, __float_as_uint(val));
    if (old == assumed) break;
  }
}

__global__ void gat_logits(const float* __restrict__ xl, const float* __restrict__ xr,
                           const float* __restrict__ att, const int* __restrict__ ei,
                           float* __restrict__ logit, float* __restrict__ mseg,
                           int H, int Cc)
{
  int idx = blockIdx.x*blockDim.x + threadIdx.x;
  if (idx >= EA_*H) return;
  int e = idx / H, h = idx - e*H;
  int s, d; edge_decode(e, ei, s, d);
  int HCd = H*Cc;
  const float* pl = xl + (size_t)s*HCd + h*Cc;
  const float* pr = xr + (size_t)d*HCd + h*Cc;
  const float* pa = att + h*Cc;
  float acc = 0.f;
  for (int c = 0; c < Cc; ++c) {
    float v = pl[c] + pr[c];
    v = (v > 0.f) ? v : NEGS_*v;
    acc += v * pa[c];
  }
  logit[idx] = acc;
  atomicMaxF(&mseg[d*H + h], acc);
}

__global__ void gat_denom(const float* __restrict__ logit, const float* __restrict__ mseg,
                          const int* __restrict__ ei, float* __restrict__ denom, int H)
{
  int idx = blockIdx.x*blockDim.x + threadIdx.x;
  if (idx >= EA_*H) return;
  int e = idx / H, h = idx - e*H;
  int s, d; edge_decode(e, ei, s, d); (void)s;
  float p = __expf(logit[idx] - mseg[d*H + h]);
  atomicAdd(&denom[d*H + h], p);
}

__global__ void gat_aggregate(const float* __restrict__ xl, const float* __restrict__ logit,
                              const float* __restrict__ mseg, const float* __restrict__ denom,
                              const int* __restrict__ ei, float* __restrict__ out,
                              int H, int Cc)
{
  int idx = blockIdx.x*blockDim.x + threadIdx.x;
  if (idx >= EA_*H) return;
  int e = idx / H, h = idx - e*H;
  int s, d; edge_decode(e, ei, s, d);
  float alpha = __expf(logit[idx] - mseg[d*H + h]) / (denom[d*H + h] + 1e-16f);
  int HCd = H*Cc;
  const float* pl = xl + (size_t)s*HCd + h*Cc;
  float* po = out + (size_t)d*HCd + h*Cc;
  for (int c = 0; c < Cc; ++c)
    atomicAdd(&po[c], pl[c]*alpha);
}

// ---------------- bias + layernorm (+relu), in place; one wave per row ----------------
__global__ void bias_ln_act(float* __restrict__ h, const float* __restrict__ bias,
                            const float* __restrict__ w, const float* __restrict__ b,
                            int M, int D, int relu)
{
  int row = blockIdx.x*(blockDim.x >> 5) + (threadIdx.x >> 5);
  if (row >= M) return;
  int lane = threadIdx.x & 31;
  int per = D >> 5;
  float* p = h + (size_t)row*D;
  float vals[8];
  float sum = 0.f;
  for (int i = 0; i < per; ++i) {
    int c = lane + i*32;
    float v = p[c] + bias[c];
    vals[i] = v; sum += v;
  }
  for (int o = 16; o > 0; o >>= 1) sum += __shfl_xor(sum, o, 32);
  float mu = sum / (float)D;
  float vs = 0.f;
  for (int i = 0; i < per; ++i) { float dlt = vals[i]-mu; vs += dlt*dlt; }
  for (int o = 16; o > 0; o >>= 1) vs += __shfl_xor(vs, o, 32);
  float rstd = rsqrtf(vs/(float)D + LN_EPS_);
  for (int i = 0; i < per; ++i) {
    int c = lane + i*32;
    float o = (vals[i]-mu)*rstd*w[c] + b[c];
    if (relu) o = fmaxf(o, 0.f);
    p[c] = o;
  }
}

// ---------------- (B,S,N,GH) -> (B*N, S, GH) ----------------
__global__ void reshape_to_seq(const float* __restrict__ h2, float* __restrict__ hs)
{
  int idx = blockIdx.x*blockDim.x + threadIdx.x;
  if (idx >= NT_*GH_) return;
  int c = idx & (GH_-1);
  int ntid = idx >> 6;
  int n = ntid % N_;
  int bs = ntid / N_;
  int s = bs % S_;
  int b = bs / S_;
  hs[(((size_t)(b*N_ + n))*S_ + s)*GH_ + c] = h2[idx];
}

// ---------------- LSTM layer ----------------
// 16 sequences per block, all 32 steps internal. Per step:
//   1. async-DMA the 16x512 xg gate-bias tile into gbuf (ASYNCcnt path)
//   2. recurrent GEMM gates = h @ whh^T with WMMA (h in LDS, K outer, 4 acc tiles/wave)
//   3. s_wait_asynccnt 0 + barrier, gbuf += acc
//   4. gate nonlinearity, update h (LDS) and c (registers)
__global__ void __launch_bounds__(256) lstm_layer(const float* __restrict__ xg,
                                                  const float* __restrict__ whh,
                                                  float* __restrict__ y,
                                                  float* __restrict__ last)
{
  __shared__ float hbuf[16*LH_];                       // 8 KB
  __shared__ __align__(16) float gbuf[16*4*LH_];       // 32 KB
  int seq0 = blockIdx.x * 16;
  int tid  = threadIdx.x;
  int wave = tid >> 5, lane = tid & 31;
  int m  = lane & 15;
  int kb = (lane >> 4) << 1;
  int n  = lane & 15;
  int mb = (lane >> 4) << 3;

  float creg[8];
  #pragma unroll
  for (int i = 0; i < 8; ++i) creg[i] = 0.f;
  for (int i = tid; i < 16*LH_; i += 256) hbuf[i] = 0.f;
  __syncthreads();

  for (int t = 0; t < S_; ++t) {
    // --- 1. async prefetch of xg tile into gbuf (overlaps with the GEMM below) ---
    #pragma unroll
    for (int r = 0; r < 8; ++r) {
      int f   = (tid + r*256) * 4;          // float index in 16x512 tile
      int sr  = f >> 9;
      int col = f & 511;
      const float* gs = xg + ((size_t)(seq0+sr)*S_ + t)*512 + col;
      uint32_t ldsa = (uint32_t)(uintptr_t)(&gbuf[f]);
      asm volatile("global_load_async_to_lds_b128 %0, %1, off"
                   :: "v"(ldsa), "v"((uint64_t)(uintptr_t)gs) : "memory");
    }

    // --- 2. gates = h @ whh^T : K outer, one A fragment feeds 4 WMMA tiles ---
    v8f acc[4];
    #pragma unroll
    for (int i = 0; i < 4; ++i) acc[i] = (v8f){};
    for (int k = 0; k < LH_; k += 4) {
      v2f a;
      a[0] = hbuf[m*LH_ + k + kb];
      a[1] = hbuf[m*LH_ + k + kb + 1];
      #pragma unroll
      for (int i = 0; i < 4; ++i) {
        int nt = wave*4 + i;
        const float* bp = whh + (size_t)(nt*16 + n)*LH_ + k + kb;  // B[k][col] = whh[col][k]
        v2f b; b[0] = bp[0]; b[1] = bp[1];
        acc[i] = __builtin_amdgcn_wmma_f32_16x16x4_f32(false, a, false, b, (short)0, acc[i], false, false);
      }
    }

    // --- 3. wait for the async tile, then gbuf = xg_tile + acc ---
    asm volatile("s_wait_asynccnt 0x0" ::: "memory");
    __syncthreads();
    #pragma unroll
    for (int i = 0; i < 4; ++i) {
      int col = (wave*4 + i)*16 + n;
      #pragma unroll
      for (int v = 0; v < 8; ++v) {
        int sr = mb + v;
        gbuf[sr*512 + col] += acc[i][v];
      }
    }
    __syncthreads();

    // --- 4. gate nonlinearity: 2048 (seq,j) cells, 8 per thread ---
    #pragma unroll
    for (int i = 0; i < 8; ++i) {
      int el = tid + i*256;
      int sr = el >> 7;
      int j  = el & (LH_-1);
      float ig = gbuf[sr*512 +           j];
      float fg = gbuf[sr*512 +   LH_ +   j];
      float gg = gbuf[sr*512 + 2*LH_ +   j];
      float og = gbuf[sr*512 + 3*LH_ +   j];
      float c  = sigm(fg)*creg[i] + sigm(ig)*tanhf(gg);
      creg[i] = c;
      float hv = sigm(og)*tanhf(c);
      hbuf[sr*LH_ + j] = hv;
      if (y)    y[((size_t)(seq0+sr)*S_ + t)*LH_ + j] = hv;
      if (last && t == S_-1) last[(size_t)(seq0+sr)*LH_ + j] = hv;
    }
    __syncthreads();
  }
}

// ---------------- FC head: one wave per row ----------------
__global__ void fc_kernel(const float* __restrict__ last, const float* __restrict__ w,
                          const float* __restrict__ b, float* __restrict__ out, int M, int K)
{
  int row = blockIdx.x*(blockDim.x >> 5) + (threadIdx.x >> 5);
  if (row >= M) return;
  int lane = threadIdx.x & 31;
  float s = 0.f;
  for (int k = lane; k < K; k += 32) s += last[(size_t)row*K + k]*w[k];
  for (int o = 16; o > 0; o >>= 1) s += __shfl_xor(s, o, 32);
  if (lane == 0) out[row] = s + b[0];
}

// ---------------- workspace layout (floats) ----------------
#define XL1_OFF   ((size_t)0)          // 6553600
#define XR1_OFF   ((size_t)6553600)    // 6553600
#define XG_OFF    ((size_t)0)          // 13107200 (aliases xl1+xr1; xg0 then xg1)
#define H1_OFF    ((size_t)13107200)   // 6553600 (out1/h1; later y0 alias)
#define Y0_OFF    ((size_t)13107200)   // 3276800
#define LOG_OFF   ((size_t)19660800)   // 1740800 (logit1, later logit2)
#define M_OFF     ((size_t)21401600)   // 102400
#define D_OFF     ((size_t)21504000)   // 102400
#define XL2_OFF   ((size_t)21606400)   // 1638400
#define XR2_OFF   ((size_t)23244800)   // 1638400
#define OUT2_OFF  ((size_t)24883200)   // 1638400
#define HSEQ_OFF  ((size_t)26521600)   // 1638400
#define LAST_OFF  ((size_t)28160000)   // 102400

static inline int gemm_blocks(int M, int N){ return ((M/16)*(N/32) + 7)/8; }

extern "C" void kernel_launch(void* const* d_in, const int* in_sizes, int n_in,
                              void* d_out, int out_size, void* d_ws, size_t ws_size,
                              hipStream_t stream) {
  const float* x        = (const float*)d_in[0];
  const int*   ei       = (const int*)  d_in[1];
  const float* g1_wl    = (const float*)d_in[2];
  const float* g1_wr    = (const float*)d_in[3];
  const float* g1_att   = (const float*)d_in[4];
  const float* g1_bias  = (const float*)d_in[5];
  const float* g2_wl    = (const float*)d_in[6];
  const float* g2_wr    = (const float*)d_in[7];
  const float* g2_att   = (const float*)d_in[8];
  const float* g2_bias  = (const float*)d_in[9];
  const float* ln1_w    = (const float*)d_in[10];
  const float* ln1_b    = (const float*)d_in[11];
  const float* ln2_w    = (const float*)d_in[12];
  const float* ln2_b    = (const float*)d_in[13];
  const float* wih0     = (const float*)d_in[14];
  const float* whh0     = (const float*)d_in[15];
  const float* bih0     = (const float*)d_in[16];
  const float* bhh0     = (const float*)d_in[17];
  const float* wih1     = (const float*)d_in[18];
  const float* whh1     = (const float*)d_in[19];
  const float* bih1     = (const float*)d_in[20];
  const float* bhh1     = (const float*)d_in[21];
  const float* fc_w     = (const float*)d_in[22];
  const float* fc_b     = (const float*)d_in[23];

  float* ws   = (float*)d_ws;
  float* xl1  = ws + XL1_OFF;
  float* xr1  = ws + XR1_OFF;
  float* xg   = ws + XG_OFF;
  float* h1   = ws + H1_OFF;
  float* y0   = ws + Y0_OFF;
  float* lg   = ws + LOG_OFF;
  float* mbuf = ws + M_OFF;
  float* dbuf = ws + D_OFF;
  float* xl2  = ws + XL2_OFF;
  float* xr2  = ws + XR2_OFF;
  float* out2 = ws + OUT2_OFF;
  float* hseq = ws + HSEQ_OFF;
  float* last = ws + LAST_OFF;
  float* pred = (float*)d_out;

  // ---- GAT layer 1 projections: (25600x32)@(32x256) ----
  wmma_gemm_f32<false><<<gemm_blocks(NT_, HC_), 256, 0, stream>>>(x, g1_wl, nullptr, nullptr, xl1, NT_, HC_, FIN_);
  wmma_gemm_f32<false><<<gemm_blocks(NT_, HC_), 256, 0, stream>>>(x, g1_wr, nullptr, nullptr, xr1, NT_, HC_, FIN_);

  // ---- init segment buffers + accumulator ----
  fill_kernel<<<(NT_*HEADS_+255)/256, 256, 0, stream>>>(mbuf, -1e30f, (size_t)NT_*HEADS_);
  fill_kernel<<<(NT_*HEADS_+255)/256, 256, 0, stream>>>(dbuf, 0.f,    (size_t)NT_*HEADS_);
  fill_kernel<<<4096, 256, 0, stream>>>(h1, 0.f, (size_t)NT_*HC_);

  // ---- GAT layer 1 edge passes ----
  int thr1 = EA_*HEADS_;
  gat_logits   <<<(thr1+255)/256, 256, 0, stream>>>(xl1, xr1, g1_att, ei, lg, mbuf, HEADS_, GH_);
  gat_denom    <<<(thr1+255)/256, 256, 0, stream>>>(lg, mbuf, ei, dbuf, HEADS_);
  gat_aggregate<<<(thr1+255)/256, 256, 0, stream>>>(xl1, lg, mbuf, dbuf, ei, h1, HEADS_, GH_);
  bias_ln_act  <<<(NT_+7)/8, 256, 0, stream>>>(h1, g1_bias, ln1_w, ln1_b, NT_, HC_, 1);

  // ---- GAT layer 2 projections: (25600x256)@(256x64) ----
  wmma_gemm_f32<false><<<gemm_blocks(NT_, GH_), 256, 0, stream>>>(h1, g2_wl, nullptr, nullptr, xl2, NT_, GH_, HC_);
  wmma_gemm_f32<false><<<gemm_blocks(NT_, GH_), 256, 0, stream>>>(h1, g2_wr, nullptr, nullptr, xr2, NT_, GH_, HC_);

  fill_kernel<<<(NT_+255)/256, 256, 0, stream>>>(mbuf, -1e30f, (size_t)NT_);
  fill_kernel<<<(NT_+255)/256, 256, 0, stream>>>(dbuf, 0.f,    (size_t)NT_);
  fill_kernel<<<4096, 256, 0, stream>>>(out2, 0.f, (size_t)NT_*GH_);

  // ---- GAT layer 2 edge passes (H=1, C=64) ----
  gat_logits   <<<(EA_+255)/256, 256, 0, stream>>>(xl2, xr2, g2_att, ei, lg, mbuf, 1, GH_);
  gat_denom    <<<(EA_+255)/256, 256, 0, stream>>>(lg, mbuf, ei, dbuf, 1);
  gat_aggregate<<<(EA_+255)/256, 256, 0, stream>>>(xl2, lg, mbuf, dbuf, ei, out2, 1, GH_);
  bias_ln_act  <<<(NT_+7)/8, 256, 0, stream>>>(out2, g2_bias, ln2_w, ln2_b, NT_, GH_, 0);

  // ---- reshape to (B*N, S, GH) ----
  reshape_to_seq<<<(NT_*GH_+255)/256, 256, 0, stream>>>(out2, hseq);

  // ---- LSTM layer 0 ----
  wmma_gemm_f32<true><<<gemm_blocks(NT_, 4*LH_), 256, 0, stream>>>(hseq, wih0, bih0, bhh0, xg, NT_, 4*LH_, GH_);
  lstm_layer<<<(B_*N_)/16, 256, 0, stream>>>(xg, whh0, y0, nullptr);

  // ---- LSTM layer 1 (only last h needed) ----
  wmma_gemm_f32<true><<<gemm_blocks(NT_, 4*LH_), 256, 0, stream>>>(y0, wih1, bih1, bhh1, xg, NT_, 4*LH_, LH_);
  lstm_layer<<<(B_*N_)/16, 256, 0, stream>>>(xg, whh1, nullptr, last);

  // ---- FC head ----
  fc_kernel<<<(B_*N_+7)/8, 256, 0, stream>>>(last, fc_w, fc_b, pred, B_*N_, LH_);
}